// SchroederReverb_1640677507495
// MI455X (gfx1250) — compile-verified
//
#include <hip/hip_runtime.h>
#include <math.h>

// Problem geometry
#define T_LEN 262144            // samples per sequence
#define N_SEQ 64                // 32 batches * 2 channels
#define NB 16                   // block length (WMMA tile dim)
#define KB (T_LEN / NB)         // 16384 blocks per sequence
#define NSEC 8                  // allpass sections
#define CH 128                  // blocks per scan chunk
#define NC (KB / CH)            // 128 chunks per sequence

typedef float v2f __attribute__((ext_vector_type(2)));
typedef float v4f __attribute__((ext_vector_type(4)));
typedef float v8f __attribute__((ext_vector_type(8)));

// Per-section derived parameters (computed on device): impulse response h,
// boundary response vectors f1/f2 (y[-1],y[-2]) and g1/g2 (x[-1],x[-2]),
// 2x2 block state transition A2 and A2^CH. 128 floats = 512B per section.
struct SecP {
  float h[16];
  float f1[16], f2[16], g1[16], g2[16];
  float A2[4];   // [[A2[0],A2[1]],[A2[2],A2[3]]]
  float A2p[4];  // A2^CH
  float pad[40];
};

// ---------------------------------------------------------------- setup ----
// atab: pre-swizzled WMMA A-operand (Toeplitz of h) per section.
// Layout: atab[((sec*4 + j)*32 + lane)] = v2f{ T[row][k0], T[row][k0+1] }
// with row = lane&15, k0 = 4*j + 2*(lane>>4)  (ISA 32-bit 16x4 A layout).
__global__ void srv_setup(const float* __restrict__ freqs,
                          const float* __restrict__ qs,
                          SecP* __restrict__ P,
                          float* __restrict__ atab,
                          unsigned* __restrict__ maxp) {
  int i = threadIdx.x;
  if (i == 0) *maxp = 0u;  // re-init every call (deterministic)
  if (i >= NSEC) return;

  const float PI = 3.14159265358979323846f;
  float w0 = 2.f * PI * freqs[i] / 48000.f;
  float alpha = sinf(w0) / (2.f * qs[i]);
  float a0 = 1.f + alpha;
  float b0 = (1.f - alpha) / a0;
  float b1 = (-2.f * cosf(w0)) / a0;
  float b2 = (1.f + alpha) / a0;
  float a1 = b1;
  float a2 = (1.f - alpha) / a0;

  SecP& sp = P[i];
  // impulse response: y[n] = b_n - a1 y[n-1] - a2 y[n-2]
  sp.h[0] = b0;
  sp.h[1] = b1 - a1 * sp.h[0];
  for (int n = 2; n < 16; ++n)
    sp.h[n] = ((n == 2) ? b2 : 0.f) - a1 * sp.h[n - 1] - a2 * sp.h[n - 2];
  // f1: response to y[-1]=1 (zero input)
  sp.f1[0] = -a1;
  sp.f1[1] = -a1 * sp.f1[0] - a2;
  for (int n = 2; n < 16; ++n) sp.f1[n] = -a1 * sp.f1[n - 1] - a2 * sp.f1[n - 2];
  // f2: response to y[-2]=1
  sp.f2[0] = -a2;
  sp.f2[1] = -a1 * sp.f2[0];
  for (int n = 2; n < 16; ++n) sp.f2[n] = -a1 * sp.f2[n - 1] - a2 * sp.f2[n - 2];
  // g1: response to x[-1]=1
  sp.g1[0] = b1;
  sp.g1[1] = b2 - a1 * sp.g1[0];
  for (int n = 2; n < 16; ++n) sp.g1[n] = -a1 * sp.g1[n - 1] - a2 * sp.g1[n - 2];
  // g2: response to x[-2]=1
  sp.g2[0] = b2;
  sp.g2[1] = -a1 * sp.g2[0];
  for (int n = 2; n < 16; ++n) sp.g2[n] = -a1 * sp.g2[n - 1] - a2 * sp.g2[n - 2];

  sp.A2[0] = sp.f1[15]; sp.A2[1] = sp.f2[15];
  sp.A2[2] = sp.f1[14]; sp.A2[3] = sp.f2[14];

  float m0 = sp.A2[0], m1 = sp.A2[1], m2 = sp.A2[2], m3 = sp.A2[3];
  for (int s = 0; s < 7; ++s) {  // A2^(2^7) = A2^128 = A2^CH
    float r0 = m0 * m0 + m1 * m2, r1 = m0 * m1 + m1 * m3;
    float r2 = m2 * m0 + m3 * m2, r3 = m2 * m1 + m3 * m3;
    m0 = r0; m1 = r1; m2 = r2; m3 = r3;
  }
  sp.A2p[0] = m0; sp.A2p[1] = m1; sp.A2p[2] = m2; sp.A2p[3] = m3;

  // Pre-swizzled, branch-free A operand table (zero-filled above diagonal).
  for (int j = 0; j < 4; ++j) {
    for (int l = 0; l < 32; ++l) {
      int row = l & 15;
      int k0 = 4 * j + 2 * (l >> 4);
      float e0 = (row >= k0)     ? sp.h[row - k0]     : 0.f;
      float e1 = (row >= k0 + 1) ? sp.h[row - k0 - 1] : 0.f;
      int base = (((i * 4) + j) * 32 + l) * 2;
      atab[base + 0] = e0;
      atab[base + 1] = e1;
    }
  }
}

// -------------------------------------------------- pass A: block drivers ---
// u_k = (partial y at block positions 15,14) incl. x-boundary terms.
__global__ void srv_passA(const float* __restrict__ xin,
                          const SecP* __restrict__ P, int sec,
                          float* __restrict__ uarr) {
  int idx = blockIdx.x * blockDim.x + threadIdx.x;  // 0 .. N_SEQ*KB-1
  int seq = idx >> 14;
  int kk = idx & (KB - 1);
  const SecP& sp = P[sec];
  const float* xb = xin + (size_t)seq * T_LEN + (size_t)kk * NB;

  const v4f* xv = (const v4f*)xb;  // 64B-aligned block
  v4f x0 = xv[0], x1 = xv[1], x2 = xv[2], x3 = xv[3];
  float xl[16];
#pragma unroll
  for (int c = 0; c < 4; ++c) {
    xl[c] = x0[c]; xl[4 + c] = x1[c]; xl[8 + c] = x2[c]; xl[12 + c] = x3[c];
  }
  float p15 = 0.f, p14 = 0.f;
#pragma unroll
  for (int d = 0; d < 16; ++d) p15 += sp.h[d] * xl[15 - d];
#pragma unroll
  for (int d = 0; d < 15; ++d) p14 += sp.h[d] * xl[14 - d];

  // Branch-free boundary samples: clamp index, mask value.
  float bm = kk ? 1.f : 0.f;
  const float* xs = xin + (size_t)seq * T_LEN;
  int t0 = kk * NB;
  int i1 = (t0 - 1 > 0) ? t0 - 1 : 0;
  int i2 = (t0 - 2 > 0) ? t0 - 2 : 0;
  float xm1 = xs[i1] * bm;
  float xm2 = xs[i2] * bm;

  v2f u;
  u[0] = p15 + sp.g1[15] * xm1 + sp.g2[15] * xm2;
  u[1] = p14 + sp.g1[14] * xm1 + sp.g2[14] * xm2;
  ((v2f*)uarr)[idx] = u;
}

// ----------------------------------------------- pass B: chunked 2x2 scan ---
__global__ void srv_scanLocal(const float* __restrict__ uarr,
                              const SecP* __restrict__ P, int sec,
                              float* __restrict__ chunkend) {
  int tid = blockIdx.x * blockDim.x + threadIdx.x;  // 0 .. N_SEQ*NC-1
  int seq = tid / NC, ch = tid % NC;
  int base = seq * KB + ch * CH;
  const SecP& sp = P[sec];
  float A0 = sp.A2[0], A1 = sp.A2[1], A2_ = sp.A2[2], A3 = sp.A2[3];
  const v2f* uv = (const v2f*)uarr;
  float w1 = 0.f, w2 = 0.f;
  for (int i = 0; i < CH; ++i) {
    v2f u = uv[base + i];
    float n1 = A0 * w1 + A1 * w2 + u[0];
    float n2 = A2_ * w1 + A3 * w2 + u[1];
    w1 = n1; w2 = n2;
  }
  v2f e; e[0] = w1; e[1] = w2;
  ((v2f*)chunkend)[tid] = e;
}

__global__ void srv_scanCarry(const float* __restrict__ chunkend,
                              const SecP* __restrict__ P, int sec,
                              float* __restrict__ carry) {
  int seq = threadIdx.x;  // 64 threads
  if (seq >= N_SEQ) return;
  const SecP& sp = P[sec];
  float A0 = sp.A2p[0], A1 = sp.A2p[1], A2_ = sp.A2p[2], A3 = sp.A2p[3];
  const v2f* ev = (const v2f*)chunkend;
  v2f* cv = (v2f*)carry;
  float c1 = 0.f, c2 = 0.f;
  for (int c = 0; c < NC; ++c) {
    int id = seq * NC + c;
    v2f cc; cc[0] = c1; cc[1] = c2;
    cv[id] = cc;
    v2f e = ev[id];
    float n1 = A0 * c1 + A1 * c2 + e[0];
    float n2 = A2_ * c1 + A3 * c2 + e[1];
    c1 = n1; c2 = n2;
  }
}

__global__ void srv_scanFinal(const float* __restrict__ uarr,
                              const SecP* __restrict__ P, int sec,
                              const float* __restrict__ carry,
                              float* __restrict__ warr) {
  int tid = blockIdx.x * blockDim.x + threadIdx.x;
  int seq = tid / NC, ch = tid % NC;
  int base = seq * KB + ch * CH;
  const SecP& sp = P[sec];
  float A0 = sp.A2[0], A1 = sp.A2[1], A2_ = sp.A2[2], A3 = sp.A2[3];
  const v2f* uv = (const v2f*)uarr;
  v2f* wv = (v2f*)warr;
  v2f c = ((const v2f*)carry)[tid];
  float w1 = c[0], w2 = c[1];
  for (int i = 0; i < CH; ++i) {
    v2f w; w[0] = w1; w[1] = w2;   // state BEFORE block: (y[16k-1], y[16k-2])
    wv[base + i] = w;
    v2f u = uv[base + i];
    float n1 = A0 * w1 + A1 * w2 + u[0];
    float n2 = A2_ * w1 + A3 * w2 + u[1];
    w1 = n1; w2 = n2;
  }
}

// ---------------------------------- pass C: WMMA block filter + boundary ----
// Each wave: 16 blocks (256 samples) of one sequence.
// D(16x16) = T(16x16 Toeplitz of h) x X(16 samples x 16 blocks),
// as 4 chained V_WMMA_F32_16X16X4_F32 (K=16 = 4*4). A operand comes from the
// pre-swizzled branch-free table; B/stores are vectorized.
__global__ __launch_bounds__(128) void srv_passC(
    const float* __restrict__ xin, float* __restrict__ xout,
    const SecP* __restrict__ P, int sec,
    const float* __restrict__ atab, const float* __restrict__ warr) {
  int wave = blockIdx.x * 4 + (threadIdx.x >> 5);
  int lane = threadIdx.x & 31;
  int seq = wave >> 10;             // KB/16 = 1024 wave-tiles per sequence
  int blk0 = (wave & 1023) << 4;    // first block of this 16-block group
  int half = lane >> 4;             // K-half selector (ISA 32-bit A/B layout)
  int col = lane & 15;              // B column N = block index within group
  const SecP& sp = P[sec];
  const float* xb = xin + (size_t)seq * T_LEN;

  // A operands (wave-uniform per section): 4 x v2f table loads, no branches.
  const v2f* At = (const v2f*)(atab) + (size_t)sec * 4 * 32 + lane;
  v2f a0 = At[0 * 32], a1 = At[1 * 32], a2 = At[2 * 32], a3 = At[3 * 32];

  // B operands: X[k0..k0+1][col] for k0 = 4j + 2*half (8B-aligned pairs).
  const float* bbase = xb + (size_t)(blk0 + col) * NB + 2 * half;
  v2f b0 = *(const v2f*)(bbase + 0);
  v2f b1 = *(const v2f*)(bbase + 4);
  v2f b2 = *(const v2f*)(bbase + 8);
  v2f b3 = *(const v2f*)(bbase + 12);

  v8f acc;
  acc = __builtin_amdgcn_wmma_f32_16x16x4_f32(false, a0, false, b0, (short)0,
                                              (v8f){0,0,0,0,0,0,0,0}, false, false);
  acc = __builtin_amdgcn_wmma_f32_16x16x4_f32(false, a1, false, b1, (short)0,
                                              acc, false, false);
  acc = __builtin_amdgcn_wmma_f32_16x16x4_f32(false, a2, false, b2, (short)0,
                                              acc, false, false);
  acc = __builtin_amdgcn_wmma_f32_16x16x4_f32(false, a3, false, b3, (short)0,
                                              acc, false, false);

  // Boundary corrections for this lane's block (col): exact IIR continuation.
  // Branch-free: clamp index, mask value (avoids EXEC save/restore pairs).
  int gb = seq * KB + blk0 + col;
  v2f w = ((const v2f*)warr)[gb];
  int t0 = (blk0 + col) * NB;
  float bm = (blk0 + col) ? 1.f : 0.f;
  int i1 = (t0 - 1 > 0) ? t0 - 1 : 0;
  int i2 = (t0 - 2 > 0) ? t0 - 2 : 0;
  float xm1 = xb[i1] * bm;
  float xm2 = xb[i2] * bm;

  // D layout: VGPR v, lanes 0-15 -> M=v ; lanes 16-31 -> M=v+8 ; N=lane&15.
  // Each lane owns 8 consecutive output samples -> two b128 stores (32B align).
  float* ob = xout + (size_t)seq * T_LEN + (size_t)t0 + 8 * half;
  v4f lo, hi;
#pragma unroll
  for (int v = 0; v < 8; ++v) {
    int m = v + 8 * half;
    float y = acc[v] + sp.f1[m] * w[0] + sp.f2[m] * w[1] +
              sp.g1[m] * xm1 + sp.g2[m] * xm2;
    y = fminf(fmaxf(y, -1.f), 1.f);  // lfilter(clamp=True) between stages
    if (v < 4) lo[v] = y; else hi[v - 4] = y;
  }
  *(v4f*)(ob + 0) = lo;
  *(v4f*)(ob + 4) = hi;
}

// ----------------------------------------- mix, global max-abs, normalize ---
__global__ void srv_reduceMax(const float* __restrict__ y8,
                              const float* __restrict__ xin,
                              unsigned* __restrict__ maxp, int n) {
  __shared__ float smax[256];
  float local = 0.f;
  for (int i = blockIdx.x * blockDim.x + threadIdx.x; i < n;
       i += gridDim.x * blockDim.x) {
    float mixed = 0.5f * (y8[i] + xin[i]);
    local = fmaxf(local, fabsf(mixed));
  }
  smax[threadIdx.x] = local;
  __syncthreads();
  for (int s = 128; s > 0; s >>= 1) {
    if (threadIdx.x < s) smax[threadIdx.x] = fmaxf(smax[threadIdx.x], smax[threadIdx.x + s]);
    __syncthreads();
  }
  if (threadIdx.x == 0)
    atomicMax(maxp, __float_as_uint(smax[0]));  // non-negative: bits ordered
}

__global__ void srv_normalize(float* __restrict__ out,
                              const float* __restrict__ xin,
                              const unsigned* __restrict__ maxp, int n) {
  int i = blockIdx.x * blockDim.x + threadIdx.x;
  if (i >= n) return;
  float mx = __uint_as_float(*maxp);
  float mixed = 0.5f * (out[i] + xin[i]);
  out[i] = mixed / mx;
}

// ------------------------------------------------------------------ launch --
extern "C" void kernel_launch(void* const* d_in, const int* in_sizes, int n_in,
                              void* d_out, int out_size, void* d_ws, size_t ws_size,
                              hipStream_t stream) {
  (void)in_sizes; (void)n_in; (void)out_size; (void)ws_size;
  const float* xin   = (const float*)d_in[0];
  const float* freqs = (const float*)d_in[1];
  const float* qs    = (const float*)d_in[2];
  float* out = (float*)d_out;

  char* ws = (char*)d_ws;
  SecP* params       = (SecP*)ws;                                  // 4 KB
  float* atab        = (float*)(ws + NSEC * sizeof(SecP));         // 8 KB
  unsigned* maxp     = (unsigned*)(ws + 16384);
  float* uarr        = (float*)(ws + 65536);                       // 8 MB
  float* warr        = uarr + 2 * (size_t)N_SEQ * KB;              // 8 MB
  float* chunkend    = warr + 2 * (size_t)N_SEQ * KB;              // 64 KB
  float* carry       = chunkend + 2 * N_SEQ * NC;                  // 64 KB
  float* sig         = carry + 2 * N_SEQ * NC;                     // 64 MB ping buf

  srv_setup<<<1, 32, 0, stream>>>(freqs, qs, params, atab, maxp);

  const float* cur = xin;
  for (int sec = 0; sec < NSEC; ++sec) {
    float* nxt = (sec & 1) ? out : sig;  // sec 7 -> d_out
    srv_passA<<<(N_SEQ * KB) / 256, 256, 0, stream>>>(cur, params, sec, uarr);
    srv_scanLocal<<<(N_SEQ * NC) / 128, 128, 0, stream>>>(uarr, params, sec, chunkend);
    srv_scanCarry<<<1, 64, 0, stream>>>(chunkend, params, sec, carry);
    srv_scanFinal<<<(N_SEQ * NC) / 128, 128, 0, stream>>>(uarr, params, sec, carry, warr);
    srv_passC<<<(N_SEQ * KB / NB) / 4, 128, 0, stream>>>(cur, nxt, params, sec, atab, warr);
    cur = nxt;
  }

  int n = N_SEQ * T_LEN;
  srv_reduceMax<<<1024, 256, 0, stream>>>(out, xin, maxp, n);
  srv_normalize<<<n / 256, 256, 0, stream>>>(out, xin, maxp, n);
}